// DirectVoxGO_23244363005990
// MI455X (gfx1250) — compile-verified
//
#include <hip/hip_runtime.h>

#define BLOCK 256
#define ITEMS 16
#define TILE  (BLOCK * ITEMS)   // 4096 elements per block

// ---------------- CDNA5 async global->LDS helpers ----------------
typedef int v4i_ __attribute__((vector_size(16)));
typedef __attribute__((address_space(1))) v4i_* as1_v4i;
typedef __attribute__((address_space(3))) v4i_* as3_v4i;

__device__ __forceinline__ void async_copy_b128(const void* gsrc, void* ldst) {
#if __has_builtin(__builtin_amdgcn_global_load_async_to_lds_b128)
  __builtin_amdgcn_global_load_async_to_lds_b128(
      (as1_v4i)(v4i_*)gsrc,      // global source (AS1, v4i32)
      (as3_v4i)(v4i_*)ldst,      // LDS destination (AS3, v4i32)
      0, 0);
#else
  // VDST = 32-bit LDS byte address (low 32 bits of generic LDS pointer),
  // VADDR = 64-bit global address.
  unsigned lds_off = (unsigned)(unsigned long long)ldst;
  asm volatile("global_load_async_to_lds_b128 %0, %1, off"
               :: "v"(lds_off), "v"(gsrc)
               : "memory");
#endif
}

__device__ __forceinline__ void wait_async0() {
#if __has_builtin(__builtin_amdgcn_s_wait_asynccnt)
  __builtin_amdgcn_s_wait_asynccnt(0);
#else
  asm volatile("s_wait_asynccnt 0" ::: "memory");
#endif
}

// ---------------- segmented-scan operator ----------------
struct Pair { float v; int f; };
__device__ __forceinline__ Pair comb(Pair a, Pair b) {
  Pair r; r.f = a.f | b.f; r.v = b.f ? b.v : (a.v + b.v); return r;
}

__device__ __forceinline__ float softplus_f(float x) {
  // numerically stable softplus
  return (x > 0.f) ? (x + log1pf(expf(-x))) : log1pf(expf(x));
}

// Stage TILE 4-byte elements from g[blockStart..] into LDS via async b128.
template <typename T>
__device__ __forceinline__ void stage_tile(const T* __restrict__ g, T* s,
                                           long long blockStart, long long Mtot) {
  const int rounds = (TILE * 4) / (BLOCK * 16);   // 4
  const int tid = threadIdx.x;
#pragma unroll
  for (int r = 0; r < rounds; ++r) {
    int c = r * BLOCK + tid;               // 16-byte chunk index within tile
    long long e = blockStart + (long long)c * 4;
    if (e > Mtot - 4) e = Mtot - 4;        // clamp: garbage-but-valid, masked later
    if (e < 0) e = 0;
    e &= ~3LL;                             // keep 16B alignment
    async_copy_b128(g + e, s + c * 4);
  }
}

// ---------------- Pass 1: per-block segmented aggregate ----------------
__global__ void __launch_bounds__(BLOCK)
k_block_agg(const float* __restrict__ density, const int* __restrict__ ray_id,
            const float* __restrict__ shiftp, const float* __restrict__ intervalp,
            float2* __restrict__ blockAgg, long long Mtot) {
  __shared__ __align__(16) float sDen[TILE];
  __shared__ __align__(16) int   sRid[TILE];
  __shared__ float sv[BLOCK];
  __shared__ int   sf[BLOCK];

  const int tid = threadIdx.x;
  const long long blockStart = (long long)blockIdx.x * TILE;

  stage_tile(density, sDen, blockStart, Mtot);
  stage_tile(ray_id,  sRid, blockStart, Mtot);
  wait_async0();
  __syncthreads();

  const float shift    = shiftp[0];
  const float interval = intervalp[0];

  const int lbase = tid * ITEMS;
  int pr;
  if (tid == 0) pr = (blockStart == 0) ? -1 : ray_id[blockStart - 1];
  else          pr = sRid[lbase - 1];

  Pair acc; acc.v = 0.f; acc.f = 0;
#pragma unroll
  for (int k = 0; k < ITEMS; ++k) {
    long long gi = blockStart + lbase + k;
    float l = 0.f; int h = 0;
    if (gi < Mtot) {
      int r = sRid[lbase + k];
      h = (r != pr);
      float x = sDen[lbase + k] + shift;
      l = -softplus_f(x) * interval;
      pr = r;
    }
    Pair e; e.v = l; e.f = h;
    acc = comb(acc, e);
  }

  sv[tid] = acc.v; sf[tid] = acc.f;
  __syncthreads();
  // ordered tree reduction (operator is associative, not commutative)
  for (int s = BLOCK / 2; s > 0; s >>= 1) {
    float va = 0.f, vb = 0.f; int fa = 0, fb = 0;
    const bool act = tid < s;
    if (act) { va = sv[2*tid]; fa = sf[2*tid]; vb = sv[2*tid+1]; fb = sf[2*tid+1]; }
    __syncthreads();
    if (act) { sf[tid] = fa | fb; sv[tid] = fb ? vb : (va + vb); }
    __syncthreads();
  }
  if (tid == 0) blockAgg[blockIdx.x] = make_float2(sf[0] ? 1.f : 0.f, sv[0]);
}

// ---------------- Pass 2: scan block aggregates (single block) ----------------
__global__ void __launch_bounds__(BLOCK)
k_scan_agg(const float2* __restrict__ agg, float2* __restrict__ carry, int NB) {
  __shared__ float sv[BLOCK];
  __shared__ int   sf[BLOCK];
  const int tid = threadIdx.x;
  const int P   = (NB + BLOCK - 1) / BLOCK;
  const int beg = tid * P;
  const int end = (beg + P < NB) ? (beg + P) : NB;

  Pair acc; acc.v = 0.f; acc.f = 0;
  for (int j = beg; j < end; ++j) {
    float2 a = agg[j];
    Pair e; e.v = a.y; e.f = (a.x != 0.f);
    acc = comb(acc, e);
  }
  sv[tid] = acc.v; sf[tid] = acc.f;
  __syncthreads();
  // ordered Hillis-Steele inclusive scan across threads
  for (int d = 1; d < BLOCK; d <<= 1) {
    float cv = sv[tid]; int cf = sf[tid];
    float pv = 0.f; int pf = 0;
    if (tid >= d) { pv = sv[tid - d]; pf = sf[tid - d]; }
    __syncthreads();
    if (tid >= d) { sv[tid] = cf ? cv : (pv + cv); sf[tid] = pf | cf; }
    __syncthreads();
  }
  Pair pre; pre.v = 0.f; pre.f = 0;
  if (tid > 0) { pre.v = sv[tid - 1]; pre.f = sf[tid - 1]; }
  for (int j = beg; j < end; ++j) {
    carry[j] = make_float2(pre.f ? 1.f : 0.f, pre.v);   // exclusive prefix
    float2 a = agg[j];
    Pair e; e.v = a.y; e.f = (a.x != 0.f);
    pre = comb(pre, e);
  }
}

// ---------------- residual transmittance init (empty rays -> 1.0) ----------------
__global__ void k_init_alphainv(float* __restrict__ a, int N) {
  int i = blockIdx.x * blockDim.x + threadIdx.x;
  if (i < N) a[i] = 1.f;
}

// ---------------- Pass 3: apply ----------------
__global__ void __launch_bounds__(BLOCK)
k_apply(const float* __restrict__ density, const int* __restrict__ ray_id,
        const float* __restrict__ shiftp, const float* __restrict__ intervalp,
        const float2* __restrict__ carry,
        float* __restrict__ weights, float* __restrict__ alphainv_last,
        long long Mtot) {
  __shared__ __align__(16) float sDen[TILE];
  __shared__ __align__(16) int   sRid[TILE];
  __shared__ float sv[BLOCK];
  __shared__ int   sf[BLOCK];

  const int tid = threadIdx.x;
  const long long blockStart = (long long)blockIdx.x * TILE;

  stage_tile(density, sDen, blockStart, Mtot);
  stage_tile(ray_id,  sRid, blockStart, Mtot);
  wait_async0();
  __syncthreads();

  const float shift    = shiftp[0];
  const float interval = intervalp[0];

  const int lbase = tid * ITEMS;
  int pr;
  if (tid == 0) pr = (blockStart == 0) ? -1 : ray_id[blockStart - 1];
  else          pr = sRid[lbase - 1];

  float lval[ITEMS];
  int hmask = 0;
  Pair acc; acc.v = 0.f; acc.f = 0;
#pragma unroll
  for (int k = 0; k < ITEMS; ++k) {
    long long gi = blockStart + lbase + k;
    float l = 0.f; int h = 0;
    if (gi < Mtot) {
      int r = sRid[lbase + k];
      h = (r != pr);
      float x = sDen[lbase + k] + shift;
      l = -softplus_f(x) * interval;
      pr = r;
    }
    lval[k] = l;
    if (h) hmask |= (1 << k);
    Pair e; e.v = l; e.f = h;
    acc = comb(acc, e);
  }

  sv[tid] = acc.v; sf[tid] = acc.f;
  __syncthreads();
  for (int d = 1; d < BLOCK; d <<= 1) {
    float cv = sv[tid]; int cf = sf[tid];
    float pv = 0.f; int pf = 0;
    if (tid >= d) { pv = sv[tid - d]; pf = sf[tid - d]; }
    __syncthreads();
    if (tid >= d) { sv[tid] = cf ? cv : (pv + cv); sf[tid] = pf | cf; }
    __syncthreads();
  }
  Pair pre; pre.v = 0.f; pre.f = 0;
  if (tid > 0) { pre.v = sv[tid - 1]; pre.f = sf[tid - 1]; }

  float2 c2 = carry[blockIdx.x];
  Pair bc; bc.v = c2.y; bc.f = (c2.x != 0.f);
  Pair cur = comb(bc, pre);   // total prefix before this thread's first item

  float w[ITEMS];
#pragma unroll
  for (int k = 0; k < ITEMS; ++k) {
    long long gi = blockStart + lbase + k;
    Pair e; e.v = lval[k]; e.f = (hmask >> k) & 1;
    cur = comb(cur, e);       // inclusive segmented sum at element gi
    if (gi < Mtot) {
      float e_exc  = cur.v - lval[k];          // exclusive within segment
      float alpha  = -expm1f(lval[k]);         // 1 - exp(l), accurate
      w[k] = expf(e_exc) * alpha;
      int r = sRid[lbase + k];
      bool last;
      if (gi == Mtot - 1) last = true;
      else {
        int nr = (lbase + k + 1 < TILE) ? sRid[lbase + k + 1] : ray_id[gi + 1];
        last = (nr != r);
      }
      if (last) alphainv_last[r] = expf(cur.v);  // one writer per ray
    } else {
      w[k] = 0.f;
    }
  }

  if (blockStart + lbase + ITEMS <= Mtot) {
    float4* wp = (float4*)(weights + blockStart + lbase);
#pragma unroll
    for (int g = 0; g < ITEMS / 4; ++g)
      wp[g] = make_float4(w[4*g+0], w[4*g+1], w[4*g+2], w[4*g+3]);
  } else {
#pragma unroll
    for (int k = 0; k < ITEMS; ++k) {
      long long gi = blockStart + lbase + k;
      if (gi < Mtot) weights[gi] = w[k];
    }
  }
}

// ---------------- host launcher ----------------
extern "C" void kernel_launch(void* const* d_in, const int* in_sizes, int n_in,
                              void* d_out, int out_size, void* d_ws, size_t ws_size,
                              hipStream_t stream) {
  const float* density   = (const float*)d_in[0];
  const float* shiftp    = (const float*)d_in[1];
  const float* intervalp = (const float*)d_in[2];
  const int*   ray_id    = (const int*)d_in[3];
  const long long M = (long long)in_sizes[0];
  const int N = out_size - (int)M;          // out = weights[M] ++ alphainv_last[N]

  float* weights  = (float*)d_out;
  float* alphainv = weights + M;

  const int NB = (int)((M + TILE - 1) / TILE);
  float2* agg   = (float2*)d_ws;
  float2* carry = agg + NB;

  k_block_agg<<<NB, BLOCK, 0, stream>>>(density, ray_id, shiftp, intervalp, agg, M);
  k_scan_agg<<<1, BLOCK, 0, stream>>>(agg, carry, NB);
  k_init_alphainv<<<(N + BLOCK - 1) / BLOCK, BLOCK, 0, stream>>>(alphainv, N);
  k_apply<<<NB, BLOCK, 0, stream>>>(density, ray_id, shiftp, intervalp, carry,
                                    weights, alphainv, M);
}